// SimilarityLossLayer_61418032333460
// MI455X (gfx1250) — compile-verified
//
#include <hip/hip_runtime.h>
#include <hip/hip_bf16.h>

// Problem constants (fixed by the reference)
#define NN   128          // N
#define TT   256          // T
#define DD   10           // D (samples feature dim)
#define EE   6            // NFILTS
#define KK   (TT*EE)      // 1536 reduction length per (i,j)

typedef __attribute__((ext_vector_type(2))) float v2f;
typedef __attribute__((ext_vector_type(8))) float v8f;

// -------- workspace layout (in floats) --------
#define WS_INVW   0                       // NN*KK       = 196608
#define WS_S      (NN*KK)                 // NN*NN       = 16384
#define WS_SECOND (WS_S + NN*NN)          // NN*NN       = 16384
#define WS_R      (WS_SECOND + NN*NN)     // NN          = 128
#define WS_PART   (WS_R + NN)             // 64
// total ~229,568 floats = ~918 KB

// ---------------------------------------------------------------------------
// K0: invw[j,k] = 1 / (err[j,t,e]^2 * 1536 * 10)   (scale of `second` folded in)
// err = input1[j, t, 7+e], input1 row stride 15.
// ---------------------------------------------------------------------------
__global__ void k_invw(const float* __restrict__ input1, float* __restrict__ invw) {
    int idx = blockIdx.x * blockDim.x + threadIdx.x;      // [0, NN*KK)
    if (idx >= NN * KK) return;
    int j = idx / KK, kk = idx - j * KK;
    int t = kk / EE,  e  = kk - t * EE;
    float err = input1[((size_t)j * TT + t) * 15 + 7 + e];
    invw[idx] = 1.0f / (err * err * (float)(KK * 10));
}

// ---------------------------------------------------------------------------
// K1: row norms r[i] = sum_d samples[i,d]^2     (one block of 128 threads)
// ---------------------------------------------------------------------------
__global__ void k_rnorm(const float* __restrict__ samples, float* __restrict__ r) {
    int i = threadIdx.x;
    float s = 0.0f;
#pragma unroll
    for (int d = 0; d < DD; ++d) { float v = samples[i * DD + d]; s += v * v; }
    r[i] = s;
}

// ---------------------------------------------------------------------------
// K2: S = (r_i + r_j - 2*G_ij)/DD  with G = samples * samples^T via
//     V_WMMA_F32_16X16X4_F32. One wave (32 threads) per 16x16 tile; 8x8 tiles.
// A-matrix 16x4 f32 layout: lanes 0-15 -> M rows, VGPR0={K0|K2}, VGPR1={K1|K3}
// (K split across lane halves). B-matrix 4x16: rows striped across lanes,
// VGPR0 = rows K0(l0-15)/K2(l16-31), VGPR1 = rows K1/K3.
// K=10 padded to 12 with zeros. Padding done with UNCONDITIONAL clamped b64
// loads + value select (no EXEC divergence: WMMA requires EXEC all-ones, and
// branchless code keeps the wave uniform throughout).
// ---------------------------------------------------------------------------
__global__ void k_gram(const float* __restrict__ samples,
                       const float* __restrict__ r,
                       float* __restrict__ S) {
    int tileM = (blockIdx.x >> 3) * 16;
    int tileN = (blockIdx.x & 7) * 16;
    int lane  = threadIdx.x;          // 0..31, full wave
    int l15   = lane & 15;
    int half  = lane >> 4;            // 0: lanes 0-15, 1: lanes 16-31
    int rowA  = tileM + l15;
    int colB  = tileN + l15;

    v8f c = {};
#pragma unroll
    for (int k = 0; k < 12; k += 4) {
        int  k0    = k + half * 2;          // {0,2} {4,6} {8,10}
        bool valid = (k0 < DD);             // pair (k0,k0+1) fully in-bounds iff k0<=8
        int  kc    = valid ? k0 : 0;        // clamp address, keep load unconditional
        float2 av = *(const float2*)(samples + rowA * DD + kc);  // 8B-aligned: even idx
        float2 bv = *(const float2*)(samples + colB * DD + kc);
        v2f a, b;
        a.x = valid ? av.x : 0.0f;  a.y = valid ? av.y : 0.0f;
        b.x = valid ? bv.x : 0.0f;  b.y = valid ? bv.y : 0.0f;
        c = __builtin_amdgcn_wmma_f32_16x16x4_f32(
                /*neg_a=*/false, a, /*neg_b=*/false, b,
                /*c_mod=*/(short)0, c, /*reuse_a=*/false, /*reuse_b=*/false);
    }

    // C/D layout: VGPR p holds M = p (lanes 0-15) or M = p+8 (lanes 16-31), N = lane%16
    float rc = r[colB];
#pragma unroll
    for (int p = 0; p < 8; ++p) {
        int row = tileM + p + half * 8;
        S[row * NN + colB] = (r[row] + rc - 2.0f * c[p]) * (1.0f / (float)DD);
    }
}

// ---------------------------------------------------------------------------
// K3: second[i*NN+j] = sum_k (y_ij[k] - yp_j[k])^2 * invw_j[k]
// One wave per (i,j) pair; contiguous float4 (global_load_b128) streaming of
// the 100.7 MB y_pred_ij tensor — this kernel is the bandwidth-bound hot spot
// (~4.3 us floor at 23.3 TB/s). yp_j / invw_j rows are reused by all 128 i's
// and stay resident in L2/WGP$.
// ---------------------------------------------------------------------------
__global__ void k_second(const float* __restrict__ ypij,
                         const float* __restrict__ ypi,
                         const float* __restrict__ invw,
                         float* __restrict__ second) {
    int wave = threadIdx.x >> 5;
    int lane = threadIdx.x & 31;
    int pair = blockIdx.x * 8 + wave;            // [0, NN*NN)
    int i = pair >> 7, j = pair & (NN - 1);

    const float4* __restrict__ y4 = (const float4*)(ypij + ((size_t)i * NN + j) * KK);
    const float4* __restrict__ p4 = (const float4*)(ypi  + (size_t)j * KK);
    const float4* __restrict__ w4 = (const float4*)(invw + (size_t)j * KK);

    float acc = 0.0f;
#pragma unroll
    for (int it = 0; it < KK / 128; ++it) {      // 12 iterations of 32 lanes x float4
        int o = it * 32 + lane;
        if (it + 1 < KK / 128) __builtin_prefetch(&y4[o + 32], 0, 0);
        float4 y = y4[o];
        float4 p = p4[o];
        float4 w = w4[o];
        float dx = y.x - p.x, dy = y.y - p.y, dz = y.z - p.z, dw = y.w - p.w;
        acc += dx * dx * w.x + dy * dy * w.y + dz * dz * w.z + dw * dw * w.w;
    }
#pragma unroll
    for (int o = 16; o > 0; o >>= 1) acc += __shfl_xor(acc, o, 32);
    if (lane == 0) second[pair] = acc;
}

// ---------------------------------------------------------------------------
// K4: per-block partial sums of |S - 0.5*(second + second^T)| (64 blocks x 256)
// ---------------------------------------------------------------------------
__global__ void k_absdiff(const float* __restrict__ S,
                          const float* __restrict__ second,
                          float* __restrict__ part) {
    __shared__ float sm[256];
    int n = blockIdx.x * 256 + threadIdx.x;      // [0, NN*NN)
    int i = n >> 7, j = n & (NN - 1);
    float v = fabsf(S[n] - 0.5f * (second[n] + second[j * NN + i]));
    sm[threadIdx.x] = v;
    __syncthreads();
#pragma unroll
    for (int s = 128; s > 0; s >>= 1) {
        if (threadIdx.x < s) sm[threadIdx.x] += sm[threadIdx.x + s];
        __syncthreads();
    }
    if (threadIdx.x == 0) part[blockIdx.x] = sm[0];
}

// ---------------------------------------------------------------------------
// K5: final sum of 64 partials -> L_sim (deterministic, no atomics, so d_out
// never needs pre-zeroing)
// ---------------------------------------------------------------------------
__global__ void k_final(const float* __restrict__ part, float* __restrict__ out) {
    __shared__ float sm[64];
    sm[threadIdx.x] = part[threadIdx.x];
    __syncthreads();
#pragma unroll
    for (int s = 32; s > 0; s >>= 1) {
        if (threadIdx.x < s) sm[threadIdx.x] += sm[threadIdx.x + s];
        __syncthreads();
    }
    if (threadIdx.x == 0) out[0] = sm[0] * (1.0f / (float)(NN * NN));
}

// ---------------------------------------------------------------------------
extern "C" void kernel_launch(void* const* d_in, const int* in_sizes, int n_in,
                              void* d_out, int out_size, void* d_ws, size_t ws_size,
                              hipStream_t stream) {
    // setup_inputs order: merged(0), y_pred_i(1), y_pred_ij(2), input1(3), samples(4), labels(5)
    const float* y_pred_i  = (const float*)d_in[1];
    const float* y_pred_ij = (const float*)d_in[2];
    const float* input1    = (const float*)d_in[3];
    const float* samples   = (const float*)d_in[4];
    float* out = (float*)d_out;

    float* ws      = (float*)d_ws;
    float* invw    = ws + WS_INVW;
    float* S       = ws + WS_S;
    float* second  = ws + WS_SECOND;
    float* r       = ws + WS_R;
    float* part    = ws + WS_PART;

    // K0: weight precompute (196608 elems)
    k_invw  <<<(NN * KK + 255) / 256, 256, 0, stream>>>(input1, invw);
    // K1: row norms
    k_rnorm <<<1, NN, 0, stream>>>(samples, r);
    // K2: WMMA Gram -> S  (8x8 tiles of 16x16, one wave each)
    k_gram  <<<64, 32, 0, stream>>>(samples, r, S);
    // K3: bandwidth-bound second[i,j] (16384 pairs, 8 waves/block)
    k_second<<<NN * NN / 8, 256, 0, stream>>>(y_pred_ij, y_pred_i, invw, second);
    // K4/K5: final reduction
    k_absdiff<<<NN * NN / 256, 256, 0, stream>>>(S, second, part);
    k_final  <<<1, 64, 0, stream>>>(part, out);
}